// objloss_65532611002444
// MI455X (gfx1250) — compile-verified
//
#include <hip/hip_runtime.h>

// ---------------------------------------------------------------------------
// YOLO-ish loss for MI455X (gfx1250, wave32).  Memory-bound: ~61MB useful
// input -> ~2.6us roofline at 23.3 TB/s.  Single streaming pass: each block
// owns a contiguous range of WHOLE batches, so per-block partials cover
// disjoint contiguous tprime intervals; the prefix sums at the last isA cell
// are reconstructed in a tiny final kernel from block partials + a <=784-cell
// recompute.  Deterministic fixed-order reductions; 32-lane sums use
// V_WMMA_F32_16X16X4_F32.
// ---------------------------------------------------------------------------

#ifndef __has_builtin
#define __has_builtin(x) 0
#endif
#if __has_builtin(__builtin_amdgcn_wmma_f32_16x16x4_f32)
#define USE_WMMA_REDUCE 1
#else
#define USE_WMMA_REDUCE 0
#endif

typedef __attribute__((ext_vector_type(2))) float v2f;
typedef __attribute__((ext_vector_type(8))) float v8f;

#define NPART 1024   // blocks in streaming pass == partials per quantity
#define BLK   256    // 8 waves of 32

// ---- wave-wide sum of 32 lane values via V_WMMA_F32_16X16X4_F32 ------------
// Step 1: A = (v,0) 16x4, B = ones(4x16)  => D[i][j] = v[i] + v[i+16]
// Step 2: feed D's 8 VGPRs back as A in 4 accumulating WMMAs; total lands in
//         every element of the accumulator.
__device__ __forceinline__ float wave_sum32(float v) {
#if USE_WMMA_REDUCE
  v2f a;    a.x = v;    a.y = 0.0f;
  v2f ones; ones.x = 1.0f; ones.y = 1.0f;
  v8f c = {};
  c = __builtin_amdgcn_wmma_f32_16x16x4_f32(false, a, false, ones, (short)0, c,
                                            false, false);
  v8f acc = {};
#pragma unroll
  for (int p = 0; p < 4; ++p) {
    v2f ap; ap.x = c[2 * p]; ap.y = c[2 * p + 1];
    acc = __builtin_amdgcn_wmma_f32_16x16x4_f32(false, ap, false, ones,
                                                (short)0, acc, false, false);
  }
  return acc[0];
#else
  for (int off = 16; off; off >>= 1) v += __shfl_xor(v, off, 32);
  return v;
#endif
}

__device__ __forceinline__ int wave_max32(int v) {
  for (int off = 16; off; off >>= 1) {
    int o = __shfl_xor(v, off, 32);
    v = (o > v) ? o : v;
  }
  return v;
}

// Deterministic block sum (result valid in thread 0 only).
__device__ __forceinline__ float block_sum(float v, float* sb) {
  float w = wave_sum32(v);
  int lane = threadIdx.x & 31, wid = threadIdx.x >> 5;
  __syncthreads();
  if (lane == 0) sb[wid] = w;
  __syncthreads();
  float t = 0.0f;
  if (threadIdx.x == 0)
    for (int i = 0; i < (int)(blockDim.x >> 5); ++i) t += sb[i];
  return t;
}

__device__ __forceinline__ int block_max(int v, int* sb) {
  int w = wave_max32(v);
  int lane = threadIdx.x & 31, wid = threadIdx.x >> 5;
  __syncthreads();
  if (lane == 0) sb[wid] = w;
  __syncthreads();
  int t = -1;
  if (threadIdx.x == 0)
    for (int i = 0; i < (int)(blockDim.x >> 5); ++i) t = (sb[i] > t) ? sb[i] : t;
  return t;
}

// ---- per-cell math, faithful to the reference ------------------------------
__device__ __forceinline__ float iou_xyxy(float ax1, float ay1, float ax2, float ay2,
                                          float bx1, float by1, float bx2, float by2) {
  bool no_ov = (ax2 < bx1) || (ax1 > bx2) || (ay2 < by1) || (ay1 > by2);
  float ix1 = fmaxf(ax1, bx1), iy1 = fmaxf(ay1, by1);
  float ix2 = fminf(ax2, bx2), iy2 = fminf(ay2, by2);
  float inter = no_ov ? 0.0f : (ix2 - ix1) * (iy2 - iy1);
  float a1 = (ax2 - ax1) * (ay2 - ay1);
  float a2 = (bx2 - bx1) * (by2 - by1);
  return (inter > 0.0f) ? inter / (a1 + a2 - inter) : 0.0f;
}

struct CellVals {
  float incr, objBm, coorc, objA;
  int tprime;
  bool isA;
};

// u walks cells in MEMORY order (b, m, n) for coalescing; tprime is the
// reference's flatten order (b, n, m) used for the last-index/prefix logic.
// Across batches, tprime order == memory order (both are b-major).
__device__ __forceinline__ void compute_cell(const float* __restrict__ pred,
                                             const float* __restrict__ lab,
                                             long long u, CellVals& o) {
  const int GG = 49;
  long long b = u / GG;
  int r  = (int)(u - b * GG);
  int mi = r / 7;            // axis -2
  int ni = r - mi * 7;       // axis -1
  const float* p = pred + b * (30LL * GG) + r;
  const float* l = lab  + b * (30LL * GG) + r;

  float p0 = p[0 * GG], p1 = p[1 * GG], p2 = p[2 * GG], p3 = p[3 * GG], p4 = p[4 * GG];
  float p5 = p[5 * GG], p6 = p[6 * GG], p7 = p[7 * GG], p8 = p[8 * GG], p9 = p[9 * GG];
  float l0 = l[0 * GG], l1 = l[1 * GG], l2 = l[2 * GG], l3 = l[3 * GG], l4 = l[4 * GG];
  float l5 = l[5 * GG], l6 = l[6 * GG], l7 = l[7 * GG], l8 = l[8 * GG];

  __builtin_prefetch(p + 2 * BLK, 0, 0);   // speculative: 2*BLK floats ahead, within this batch's planes
  __builtin_prefetch(l + 2 * BLK, 0, 0);

  const float inv7 = 1.0f / 7.0f;
  float fm = (float)mi, fn = (float)ni;

  float x1 = (p0 + fm) * inv7, y1 = (p1 + fn) * inv7;
  float b1x1 = x1 - 0.5f * p2, b1y1 = y1 - 0.5f * p3;
  float b1x2 = x1 + 0.5f * p2, b1y2 = y1 + 0.5f * p3;

  float x2 = (p5 + fm) * inv7, y2 = (p6 + fn) * inv7;
  float b2x1 = x2 - 0.5f * p7, b2y1 = y2 - 0.5f * p8;
  float b2x2 = x2 + 0.5f * p7, b2y2 = y2 + 0.5f * p8;

  float xg = (l0 + fm) * inv7, yg = (l1 + fn) * inv7;
  float gx1 = xg - 0.5f * l2, gy1 = yg - 0.5f * l3;
  float gx2 = xg + 0.5f * l2, gy2 = yg + 0.5f * l3;

  float iou1 = iou_xyxy(b1x1, b1y1, b1x2, b1y2, gx1, gy1, gx2, gy2);
  float iou2 = iou_xyxy(b2x1, b2y1, b2x2, b2y2, gx1, gy1, gx2, gy2);

  float d0 = p0 - l0, d1 = p1 - l1;
  float s2 = sqrtf(p2) - sqrtf(l2), s3 = sqrtf(p3) - sqrtf(l3);
  float coorA = 5.0f * (d0 * d0 + d1 * d1 + s2 * s2 + s3 * s3);
  float d5 = p5 - l5, d6 = p6 - l6;
  float s7 = sqrtf(p7) - sqrtf(l7), s8 = sqrtf(p8) - sqrtf(l8);
  float coorB = 5.0f * (d5 * d5 + d6 * d6 + s7 * s7 + s8 * s8);

  float oA = p4 - iou1; oA *= oA;
  float oB = p9 - iou2; oB *= oB;
  float noobj_bg = 0.5f * (p4 * p4 + p9 * p9);

  bool mask = (l4 == 1.0f);
  bool selA = (iou1 >= iou2);

  o.incr   = mask ? (selA ? 0.5f * oB : 0.5f * oA) : noobj_bg;
  o.objBm  = (mask && !selA) ? oB : 0.0f;
  o.coorc  = mask ? (selA ? coorA : coorB) : 0.0f;
  o.objA   = oA;
  o.isA    = mask && selA;
  o.tprime = (int)(b * (long long)GG + ni * 7 + mi);
}

// ---- single streaming pass: block k owns batches [k*bpb, (k+1)*bpb) --------
__global__ void loss_pass(const float* __restrict__ pred,
                          const float* __restrict__ lab,
                          long long nbatch, int bpb,
                          float* __restrict__ pIncr, float* __restrict__ pObjB,
                          float* __restrict__ pCoor, int* __restrict__ pMax) {
  long long b0 = (long long)blockIdx.x * bpb;
  long long b1 = b0 + bpb; if (b1 > nbatch) b1 = nbatch;
  long long cells = (b1 > b0) ? (b1 - b0) * 49 : 0;
  long long base  = b0 * 49;

  float sI = 0.0f, sB = 0.0f, sC = 0.0f;
  int mIdx = -1;
  for (long long j = threadIdx.x; j < cells; j += BLK) {
    CellVals c;
    compute_cell(pred, lab, base + j, c);
    sI += c.incr; sB += c.objBm; sC += c.coorc;
    if (c.isA && c.tprime > mIdx) mIdx = c.tprime;
  }

  __shared__ float sbf[BLK / 32];
  __shared__ int   sbi[BLK / 32];
  float tI = block_sum(sI, sbf);
  float tB = block_sum(sB, sbf);
  float tC = block_sum(sC, sbf);
  int   tM = block_max(mIdx, sbi);
  if (threadIdx.x == 0) {
    pIncr[blockIdx.x] = tI;
    pObjB[blockIdx.x] = tB;
    pCoor[blockIdx.x] = tC;
    pMax[blockIdx.x]  = tM;
  }
}

// ---- final: combine partials, reconstruct prefix sums at lastA, finalize ---
__global__ void loss_final(const float* __restrict__ pred,
                           const float* __restrict__ lab,
                           const float* __restrict__ pIncr,
                           const float* __restrict__ pObjB,
                           const float* __restrict__ pCoor,
                           const int* __restrict__ pMax,
                           int bpb, float inv_nbatch, float* __restrict__ out) {
  // Phase 1: global reductions over NPART block partials.
  float a = 0.0f, bb = 0.0f, cc = 0.0f;
  int m = -1;
  for (int i = threadIdx.x; i < NPART; i += BLK) {
    a += pIncr[i]; bb += pObjB[i]; cc += pCoor[i];
    int pm = pMax[i]; m = (pm > m) ? pm : m;
  }
  __shared__ float sbf[BLK / 32];
  __shared__ int   sbi[BLK / 32];
  float noobj = block_sum(a, sbf);
  float totB  = block_sum(bb, sbf);
  float coor  = block_sum(cc, sbf);
  int   lastA = block_max(m, sbi);

  // Broadcast scalars block-wide (keeps later branch uniform -> EXEC all-1s
  // at every WMMA, as the ISA requires).
  __shared__ float shS[3];
  __shared__ int   shLast;
  if (threadIdx.x == 0) { shS[0] = noobj; shS[1] = totB; shS[2] = coor; shLast = lastA; }
  __syncthreads();
  noobj = shS[0]; totB = shS[1]; coor = shS[2]; lastA = shLast;

  float obj = totB;  // value when no isA cell exists
  if (lastA >= 0) {
    long long bLast = (long long)lastA / 49;
    int kb = (int)(bLast / bpb);  // streaming block that owned batch bLast

    // Sum of whole-block partials strictly below kb (tprime-contiguous).
    float aBelow = 0.0f, bBelow = 0.0f;
    for (int i = threadIdx.x; i < kb; i += BLK) {
      aBelow += pIncr[i]; bBelow += pObjB[i];
    }

    // Recompute block kb's cells for batches [kb*bpb, bLast] (<= bpb*49).
    __shared__ float shObjA;
    if (threadIdx.x == 0) shObjA = 0.0f;
    __syncthreads();
    long long b0 = (long long)kb * bpb;
    long long cells = (bLast - b0 + 1) * 49;
    long long base  = b0 * 49;
    float aIn = 0.0f, bIn = 0.0f;
    for (long long j = threadIdx.x; j < cells; j += BLK) {
      CellVals c;
      compute_cell(pred, lab, base + j, c);
      if (c.tprime <  lastA) aIn += c.incr;    // exclusive prefix of incr
      if (c.tprime <= lastA) bIn += c.objBm;   // inclusive prefix of objB_masked
      if (c.tprime == lastA) shObjA = c.objA;  // unique writer
    }
    __syncthreads();
    float incrBefore = block_sum(aBelow + aIn, sbf);
    float objBUpto   = block_sum(bBelow + bIn, sbf);
    obj = incrBefore + shObjA + (totB - objBUpto);  // thread0-valid
  }

  if (threadIdx.x == 0)
    out[0] = (2.0f * coor + obj + noobj) * inv_nbatch;
}

// ---------------------------------------------------------------------------
extern "C" void kernel_launch(void* const* d_in, const int* in_sizes, int n_in,
                              void* d_out, int out_size, void* d_ws, size_t ws_size,
                              hipStream_t stream) {
  const float* pred = (const float*)d_in[0];
  const float* lab  = (const float*)d_in[1];
  long long ncells  = (long long)in_sizes[0] / 30;   // B * 49
  long long nbatch  = ncells / 49;
  int bpb           = (int)((nbatch + NPART - 1) / NPART);  // whole batches/block
  float inv_nbatch  = 1.0f / (float)nbatch;

  float* wsf = (float*)d_ws;
  int*   wsi = (int*)d_ws;
  float* pIncr = wsf + 0;
  float* pObjB = wsf + NPART;
  float* pCoor = wsf + 2 * NPART;
  int*   pMax  = wsi + 3 * NPART;

  loss_pass<<<NPART, BLK, 0, stream>>>(pred, lab, nbatch, bpb,
                                       pIncr, pObjB, pCoor, pMax);
  loss_final<<<1, BLK, 0, stream>>>(pred, lab, pIncr, pObjB, pCoor, pMax,
                                    bpb, inv_nbatch, (float*)d_out);
}